// LightCLHHN_76596446757446
// MI455X (gfx1250) — compile-verified
//
#include <hip/hip_runtime.h>
#include <hip/hip_bf16.h>
#include <math.h>

// ---------------- problem constants ----------------
constexpr int B = 512, L = 50, N = 64, E = 16, H = 128, D = 256;
constexpr int ITEM_NUM = 100000;
constexpr float EPS = 1e-12f;

typedef __attribute__((ext_vector_type(2))) float v2f;
typedef __attribute__((ext_vector_type(8))) float v8f;

__device__ __forceinline__ float wave_sum32(float v) {
  for (int o = 16; o > 0; o >>= 1) v += __shfl_xor(v, o, 32);
  return v;
}

// ================= Tensor Data Mover staging (probe-verified 5-arg form) =================
#if __has_builtin(__builtin_amdgcn_tensor_load_to_lds) && (__clang_major__ < 23)
#define HAVE_TDM 1
typedef __attribute__((ext_vector_type(4))) unsigned int u32x4;
typedef __attribute__((ext_vector_type(8))) int i32x8;
typedef __attribute__((ext_vector_type(4))) int i32x4;

// 2-D f32 tile load: tensor row length 256 f32 (stride 256), tile 256 x tile_rows,
// LDS padding: +4 DWORDs every 256 DWORDs -> padded LDS row stride = 260 floats.
// Rows beyond tensor_rows zero-fill (hardware OOB semantics).
__device__ __forceinline__ void tdm_load_2d(unsigned lds_byte_addr, const float* gaddr,
                                            unsigned tensor_rows, unsigned tile_rows) {
  unsigned long long ga = (unsigned long long)gaddr;
  u32x4 g0;
  g0[0] = 0x1u;                                  // count=1 (valid), user mode
  g0[1] = lds_byte_addr;                         // lds_addr (bytes)
  g0[2] = (unsigned)(ga & 0xFFFFFFFFu);          // global_addr[31:0]
  g0[3] = (unsigned)(ga >> 32) | (2u << 30);     // global_addr[56:32] | type=2
  i32x8 g1;
  g1[0] = (int)0x07D20000u;  // data_size=4B | pad_enable | pad_interval=256dw | pad_amount=4dw
  g1[1] = (int)(256u << 16);                     // tensor_dim0 = 256 (low 16)
  g1[2] = (int)((tensor_rows & 0xFFFFu) << 16);  // dim0 hi(0) | tensor_dim1 lo
  g1[3] = (int)((tensor_rows >> 16) | (256u << 16)); // tensor_dim1 hi | tile_dim0=256
  g1[4] = (int)(tile_rows & 0xFFFFu);            // tile_dim1 | tile_dim2=0
  g1[5] = 256;                                   // tensor_dim0_stride lo
  g1[6] = 0;                                     // stride hi | dim1_stride lo (2-D: unused)
  g1[7] = 0;
  i32x4 zz = {0, 0, 0, 0};                       // groups 2/3 unused (2-D tensor)
  __builtin_amdgcn_tensor_load_to_lds(g0, g1, zz, zz, 0);
}
#else
#define HAVE_TDM 0
#endif

__device__ __forceinline__ unsigned lds_byte_off(const void* p) {
  // generic LDS pointer keeps the LDS offset in addr[31:0] (aperture in high bits)
  return (unsigned)(unsigned long long)p;
}

// ---------------- K1: nodes_hidden = l2norm(embedding[nodes]) ----------------
__global__ __launch_bounds__(256) void k_nodes_init(const float* __restrict__ emb,
                                                    const int* __restrict__ nodes,
                                                    float* __restrict__ n_h) {
  int gid  = blockIdx.x * 256 + threadIdx.x;
  int row  = gid >> 5;
  int lane = gid & 31;
  if (row >= B * N) return;
  int idx = nodes[row];
  const float* src = emb + (size_t)idx * H + lane * 4;
  float a = src[0], b = src[1], c = src[2], d = src[3];
  float ss = wave_sum32(a * a + b * b + c * c + d * d);
  float inv = 1.0f / fmaxf(sqrtf(ss), EPS);
  float* dst = n_h + (size_t)row * H + lane * 4;
  dst[0] = a * inv; dst[1] = b * inv; dst[2] = c * inv; dst[3] = d * inv;
}

// ---------------- K2a: e_tmp[b,e,:] = (sum_n adj[b,n,e]*n_h[b,n,:]) / deg_e ----------------
__global__ __launch_bounds__(256) void k_edge_agg(const float* __restrict__ adj,
                                                  const float* __restrict__ n_h,
                                                  float* __restrict__ e_tmp) {
  int gid  = blockIdx.x * 256 + threadIdx.x;
  int row  = gid >> 5;
  int lane = gid & 31;
  if (row >= B * E) return;
  int b = row >> 4, e = row & 15;
  const float* a  = adj + (size_t)b * N * E + e;
  const float* nh = n_h + (size_t)b * N * H + lane * 4;
  float s0 = 0, s1 = 0, s2 = 0, s3 = 0, deg = 0;
  for (int n = 0; n < N; ++n) {
    float w = a[n * E];
    deg += w;
    if (w != 0.0f) {
      const float* p = nh + (size_t)n * H;
      s0 += w * p[0]; s1 += w * p[1]; s2 += w * p[2]; s3 += w * p[3];
    }
  }
  float inv = 1.0f / fmaxf(deg, 1.0f);
  float* dst = e_tmp + (size_t)row * H + lane * 4;
  dst[0] = s0 * inv; dst[1] = s1 * inv; dst[2] = s2 * inv; dst[3] = s3 * inv;
}

// ---------------- K3a: n_tmp[b,n,:] = (sum_e adj[b,n,e]*e_h[b,e,:]) / deg_n ----------------
__global__ __launch_bounds__(256) void k_node_agg(const float* __restrict__ adj,
                                                  const float* __restrict__ e_h,
                                                  float* __restrict__ n_tmp) {
  int gid  = blockIdx.x * 256 + threadIdx.x;
  int row  = gid >> 5;
  int lane = gid & 31;
  if (row >= B * N) return;
  int b = row >> 6, n = row & 63;
  const float* a  = adj + (size_t)b * N * E + (size_t)n * E;
  const float* eh = e_h + (size_t)b * E * H + lane * 4;
  float s0 = 0, s1 = 0, s2 = 0, s3 = 0, deg = 0;
  for (int e = 0; e < E; ++e) {
    float w = a[e];
    deg += w;
    if (w != 0.0f) {
      const float* p = eh + (size_t)e * H;
      s0 += w * p[0]; s1 += w * p[1]; s2 += w * p[2]; s3 += w * p[3];
    }
  }
  float inv = 1.0f / fmaxf(deg, 1.0f);
  float* dst = n_tmp + (size_t)row * H + lane * 4;
  dst[0] = s0 * inv; dst[1] = s1 * inv; dst[2] = s2 * inv; dst[3] = s3 * inv;
}

// ---------------- K2b/K3b: out[r,:] = in[r,:] @ W (1x128 @ 128x128) ----------------
__global__ __launch_bounds__(128) void k_row_matmul128(const float* __restrict__ in,
                                                       const float* __restrict__ W,
                                                       float* __restrict__ out) {
  __shared__ float x[H];
  int r = blockIdx.x, t = threadIdx.x;
  x[t] = in[(size_t)r * H + t];
  __syncthreads();
  float acc = 0.0f;
#pragma unroll 8
  for (int k = 0; k < H; ++k) acc += x[k] * W[k * H + t];
  out[(size_t)r * H + t] = acc;
}

// ---------------- K4: seq_hidden = l2norm(concat(item_gather, cate_gather)) ----------------
__global__ __launch_bounds__(256) void k_seq_build(const float* __restrict__ n_h,
                                                   const int* __restrict__ alias_item,
                                                   const int* __restrict__ alias_cate,
                                                   float* __restrict__ seq) {
  int gid  = blockIdx.x * 256 + threadIdx.x;
  int row  = gid >> 5;
  int lane = gid & 31;
  if (row >= B * L) return;
  int b = row / L;
  int ai = alias_item[row], ac = alias_cate[row];
  const float* pi = n_h + (size_t)b * N * H + (size_t)ai * H + lane * 4;
  const float* pc = n_h + (size_t)b * N * H + (size_t)ac * H + lane * 4;
  float i0 = pi[0], i1 = pi[1], i2 = pi[2], i3 = pi[3];
  float c0 = pc[0], c1 = pc[1], c2 = pc[2], c3 = pc[3];
  float ss = wave_sum32(i0*i0 + i1*i1 + i2*i2 + i3*i3 + c0*c0 + c1*c1 + c2*c2 + c3*c3);
  float inv = 1.0f / fmaxf(sqrtf(ss), EPS);
  float* dst = seq + (size_t)row * D;
  float* d0 = dst + lane * 4;
  float* d1 = dst + H + lane * 4;
  d0[0] = i0 * inv; d0[1] = i1 * inv; d0[2] = i2 * inv; d0[3] = i3 * inv;
  d1[0] = c0 * inv; d1[1] = c1 * inv; d1[2] = c2 * inv; d1[3] = c3 * inv;
}

// ---------------- K5a: q1[b,:] = ht @ q1_w,  ht = seq[b, last_idx[b], :] ----------------
__global__ __launch_bounds__(256) void k_q1(const int* __restrict__ item_seq,
                                            const float* __restrict__ seq,
                                            const float* __restrict__ q1_w,
                                            float* __restrict__ q1) {
  __shared__ float ht[D];
  __shared__ int lidx;
  int b = blockIdx.x, t = threadIdx.x;
  if (t == 0) {
    int c = 0;
    for (int l = 0; l < L; ++l) c += (item_seq[b * L + l] > 0) ? 1 : 0;
    lidx = max(c - 1, 0);
  }
  __syncthreads();
  ht[t] = seq[((size_t)b * L + lidx) * D + t];
  __syncthreads();
  float acc = 0.0f;
#pragma unroll 8
  for (int k = 0; k < D; ++k) acc += ht[k] * q1_w[k * D + t];
  q1[(size_t)b * D + t] = acc;
}

// ---------------- K_q2: q2 = seq @ q2_w via f32 WMMA ----------------
// block = 256 thr (8 waves); tile M=128 x N=64; each wave: 16 rows x 64 cols (4 accs)
constexpr int LDT = 260;   // padded K-major row stride (260 % 64 == 4 -> conflict-free)
constexpr int LDB2 = 68;   // padded N-major row stride for row-major B tile
__global__ __launch_bounds__(256) void k_q2_wmma(const float* __restrict__ seq,
                                                 const float* __restrict__ q2_w,
                                                 float* __restrict__ q2) {
  __shared__ float lA[128 * LDT];   // 133,120 B
  __shared__ float lB[256 * LDB2];  //  69,632 B
  const int m0 = blockIdx.x * 128;  // 25600/128 = 200 exact
  const int n0 = blockIdx.y * 64;   // 256/64 = 4 exact
  const int tid = threadIdx.x;

  for (int i = tid; i < 128 * 64; i += 256) {       // A: 128 x 256
    int r = i >> 6, c = (i & 63) * 4;
    const float* s = seq + ((size_t)(m0 + r)) * D + c;
    float* d = lA + r * LDT + c;
    d[0] = s[0]; d[1] = s[1]; d[2] = s[2]; d[3] = s[3];
  }
  for (int i = tid; i < 256 * 16; i += 256) {       // B: 256 x 64 (row-major k x n)
    int r = i >> 4, c = (i & 15) * 4;
    const float* s = q2_w + (size_t)r * D + n0 + c;
    float* d = lB + r * LDB2 + c;
    d[0] = s[0]; d[1] = s[1]; d[2] = s[2]; d[3] = s[3];
  }
  __syncthreads();

  const int wave = tid >> 5, lane = tid & 31;
  const int idx16 = lane & 15, koff = (lane >> 4) * 2;
  const float* pa = lA + (wave * 16 + idx16) * LDT + koff;
  v8f a0 = {}, a1 = {}, a2 = {}, a3 = {};
  for (int k = 0; k < D; k += 4) {
    v2f a; a.x = pa[k]; a.y = pa[k + 1];
    const float* pb0 = lB + (k + koff) * LDB2 + idx16;
    const float* pb1 = pb0 + LDB2;
    v2f b;
    b.x = pb0[0];  b.y = pb1[0];
    a0 = __builtin_amdgcn_wmma_f32_16x16x4_f32(false, a, false, b, (short)0, a0, false, false);
    b.x = pb0[16]; b.y = pb1[16];
    a1 = __builtin_amdgcn_wmma_f32_16x16x4_f32(false, a, false, b, (short)0, a1, false, false);
    b.x = pb0[32]; b.y = pb1[32];
    a2 = __builtin_amdgcn_wmma_f32_16x16x4_f32(false, a, false, b, (short)0, a2, false, false);
    b.x = pb0[48]; b.y = pb1[48];
    a3 = __builtin_amdgcn_wmma_f32_16x16x4_f32(false, a, false, b, (short)0, a3, false, false);
  }
  const int mrow = m0 + wave * 16 + (lane >> 4) * 8;
  const int ncol = n0 + idx16;
#pragma unroll
  for (int i = 0; i < 8; ++i) {
    float* o = q2 + (size_t)(mrow + i) * D + ncol;
    o[0]  = a0[i];
    o[16] = a1[i];
    o[32] = a2[i];
    o[48] = a3[i];
  }
}

// ---------------- K5b: alpha[b,l] = sigmoid(q1[b]+q2[b,l]+q1_b) . v_w ----------------
__global__ __launch_bounds__(256) void k_alpha(const float* __restrict__ q2,
                                               const float* __restrict__ q1,
                                               const float* __restrict__ q1_b,
                                               const float* __restrict__ v_w,
                                               float* __restrict__ alpha) {
  int gid  = blockIdx.x * 256 + threadIdx.x;
  int row  = gid >> 5;
  int lane = gid & 31;
  if (row >= B * L) return;
  int b = row / L;
  float s = 0.0f;
#pragma unroll
  for (int j = 0; j < 8; ++j) {
    int t = lane * 8 + j;
    float z = q1[(size_t)b * D + t] + q2[(size_t)row * D + t] + q1_b[t];
    s += v_w[t] / (1.0f + expf(-z));
  }
  s = wave_sum32(s);
  if (lane == 0) alpha[row] = s;
}

// ---------------- K5c: qn[b,:] = l2norm(sum_l alpha*mask*seq) ----------------
__global__ __launch_bounds__(256) void k_seqout(const float* __restrict__ seq,
                                                const float* __restrict__ alpha,
                                                const int* __restrict__ item_seq,
                                                float* __restrict__ qn) {
  __shared__ float red[256];
  int b = blockIdx.x, t = threadIdx.x;
  float acc = 0.0f;
  for (int l = 0; l < L; ++l) {
    if (item_seq[b * L + l] > 0)
      acc += alpha[b * L + l] * seq[((size_t)b * L + l) * D + t];
  }
  red[t] = acc * acc;
  __syncthreads();
  for (int s = 128; s > 0; s >>= 1) {
    if (t < s) red[t] += red[t + s];
    __syncthreads();
  }
  __syncthreads();
  float inv = 1.0f / fmaxf(sqrtf(red[0]), EPS);
  qn[(size_t)b * D + t] = acc * inv;
}

// ---------------- K6: item_norm[i,:] = l2norm(concat(emb[i], emb[item_cates[i]])) ----------------
__global__ __launch_bounds__(256) void k_item_norm(const float* __restrict__ emb,
                                                   const int* __restrict__ item_cates,
                                                   float* __restrict__ itn) {
  int gid  = blockIdx.x * 256 + threadIdx.x;
  int row  = gid >> 5;
  int lane = gid & 31;
  if (row >= ITEM_NUM) return;
  const float* e1 = emb + (size_t)row * H + lane * 4;
  const float* e2 = emb + (size_t)item_cates[row] * H + lane * 4;
  float a0 = e1[0], a1 = e1[1], a2 = e1[2], a3 = e1[3];
  float c0 = e2[0], c1 = e2[1], c2 = e2[2], c3 = e2[3];
  float ss = wave_sum32(a0*a0 + a1*a1 + a2*a2 + a3*a3 + c0*c0 + c1*c1 + c2*c2 + c3*c3);
  float inv = 1.0f / fmaxf(sqrtf(ss), EPS);
  float* dst = itn + (size_t)row * D;
  float* d0 = dst + lane * 4;
  float* d1 = dst + H + lane * 4;
  d0[0] = a0 * inv; d0[1] = a1 * inv; d0[2] = a2 * inv; d0[3] = a3 * inv;
  d1[0] = c0 * inv; d1[1] = c1 * inv; d1[2] = c2 * inv; d1[3] = c3 * inv;
}

// ---------------- K7: scores = 16 * qn @ itn^T via f32 WMMA + TDM staging ----------------
// block = 256 thr (8 waves); tile M=64 x N=128; wave (wm,wn): rows wm*16..+16, cols wn*64..+64
__global__ __launch_bounds__(256) void k_scores_wmma(const float* __restrict__ qn,
                                                     const float* __restrict__ itn,
                                                     float* __restrict__ out) {
  __shared__ float lA[64 * LDT];    //  66,560 B
  __shared__ float lB[128 * LDT];   // 133,120 B
  const int m0  = blockIdx.x * 64;   // grid.x = 8 (512/64)
  const int n0  = blockIdx.y * 128;  // grid.y = 782 (ceil 100000/128)
  const int tid = threadIdx.x;

#if HAVE_TDM
  if (tid < 32) {  // wave 0 issues both DMA descriptors
    tdm_load_2d(lds_byte_off(lA), qn  + (size_t)m0 * D, (unsigned)(B - m0), 64u);
    tdm_load_2d(lds_byte_off(lB), itn + (size_t)n0 * D, (unsigned)(ITEM_NUM - n0), 128u);
    __builtin_amdgcn_s_wait_tensorcnt(0);
  }
  __syncthreads();
#else
  for (int i = tid; i < 64 * 64; i += 256) {        // A: 64 x 256
    int r = i >> 6, c = (i & 63) * 4;
    const float* s = qn + ((size_t)(m0 + r)) * D + c;
    float* d = lA + r * LDT + c;
    d[0] = s[0]; d[1] = s[1]; d[2] = s[2]; d[3] = s[3];
  }
  for (int i = tid; i < 128 * 64; i += 256) {       // B: 128 x 256 (zero-pad OOB items)
    int r = i >> 6, c = (i & 63) * 4;
    int item = n0 + r;
    float* d = lB + r * LDT + c;
    if (item < ITEM_NUM) {
      const float* s = itn + (size_t)item * D + c;
      d[0] = s[0]; d[1] = s[1]; d[2] = s[2]; d[3] = s[3];
    } else {
      d[0] = 0.f; d[1] = 0.f; d[2] = 0.f; d[3] = 0.f;
    }
  }
  __syncthreads();
#endif

  const int wave = tid >> 5, lane = tid & 31;
  const int wm = wave & 3, wn = wave >> 2;
  const int idx16 = lane & 15, koff = (lane >> 4) * 2;
  const float* pa = lA + (wm * 16 + idx16) * LDT + koff;
  const float* pb = lB + (wn * 64 + idx16) * LDT + koff;

  v8f a0 = {}, a1 = {}, a2 = {}, a3 = {};
  for (int k = 0; k < D; k += 4) {
    v2f a; a.x = pa[k]; a.y = pa[k + 1];
    v2f b;
    b.x = pb[k];            b.y = pb[k + 1];
    a0 = __builtin_amdgcn_wmma_f32_16x16x4_f32(false, a, false, b, (short)0, a0, false, false);
    b.x = pb[16 * LDT + k]; b.y = pb[16 * LDT + k + 1];
    a1 = __builtin_amdgcn_wmma_f32_16x16x4_f32(false, a, false, b, (short)0, a1, false, false);
    b.x = pb[32 * LDT + k]; b.y = pb[32 * LDT + k + 1];
    a2 = __builtin_amdgcn_wmma_f32_16x16x4_f32(false, a, false, b, (short)0, a2, false, false);
    b.x = pb[48 * LDT + k]; b.y = pb[48 * LDT + k + 1];
    a3 = __builtin_amdgcn_wmma_f32_16x16x4_f32(false, a, false, b, (short)0, a3, false, false);
  }

  // D layout: VGPR i -> M = i + (lane>=16 ? 8 : 0); N = lane & 15
  const int mbase = m0 + wm * 16 + (lane >> 4) * 8;
  const int nb = n0 + wn * 64 + idx16;
#pragma unroll
  for (int j = 0; j < 4; ++j) {
    int ncol = nb + j * 16;
    if (ncol < ITEM_NUM) {
      const v8f& acc = (j == 0) ? a0 : (j == 1) ? a1 : (j == 2) ? a2 : a3;
#pragma unroll
      for (int i = 0; i < 8; ++i)
        out[(size_t)(mbase + i) * ITEM_NUM + ncol] = acc[i] * 16.0f;
    }
  }
}

// ---------------- orchestration ----------------
extern "C" void kernel_launch(void* const* d_in, const int* in_sizes, int n_in,
                              void* d_out, int out_size, void* d_ws, size_t ws_size,
                              hipStream_t stream) {
  const int*   item_seq   = (const int*)  d_in[0];
  // d_in[1] = label (unused by forward scores)
  const int*   nodes      = (const int*)  d_in[2];
  const float* hn_adj     = (const float*)d_in[3];
  const int*   alias_item = (const int*)  d_in[4];
  const int*   alias_cate = (const int*)  d_in[5];
  const float* embedding  = (const float*)d_in[6];
  const int*   item_cates = (const int*)  d_in[7];
  const float* w_edge     = (const float*)d_in[8];
  const float* w_node     = (const float*)d_in[9];
  const float* q1_w       = (const float*)d_in[10];
  const float* q1_b       = (const float*)d_in[11];
  const float* q2_w       = (const float*)d_in[12];
  const float* v_w        = (const float*)d_in[13];
  float* scores = (float*)d_out;

  // workspace layout (floats, all float4-aligned)
  float* ws    = (float*)d_ws;
  float* n_h   = ws;                         // B*N*H   = 4,194,304
  float* e_tmp = n_h   + (size_t)B * N * H;  // B*E*H   = 1,048,576
  float* e_h   = e_tmp + (size_t)B * E * H;  // B*E*H   = 1,048,576
  float* n_tmp = e_h   + (size_t)B * E * H;  // B*N*H   = 4,194,304
  float* seq   = n_tmp + (size_t)B * N * H;  // B*L*D   = 6,553,600
  float* q2b   = seq   + (size_t)B * L * D;  // B*L*D   = 6,553,600
  float* q1    = q2b   + (size_t)B * L * D;  // B*D     =   131,072
  float* alpha = q1    + (size_t)B * D;      // B*L     =    25,600
  float* qn    = alpha + (size_t)B * L;      // B*D     =   131,072
  float* itn   = qn    + (size_t)B * D;      // ITEM*D  = 25,600,000

  // 1) gather + normalize node embeddings
  k_nodes_init<<<(B * N * 32) / 256, 256, 0, stream>>>(embedding, nodes, n_h);

  // 2) HGNN propagation, STEP = 2
  for (int step = 0; step < 2; ++step) {
    k_edge_agg<<<(B * E * 32) / 256, 256, 0, stream>>>(hn_adj, n_h, e_tmp);
    k_row_matmul128<<<B * E, 128, 0, stream>>>(e_tmp, w_edge, e_h);
    k_node_agg<<<(B * N * 32) / 256, 256, 0, stream>>>(hn_adj, e_h, n_tmp);
    k_row_matmul128<<<B * N, 128, 0, stream>>>(n_tmp, w_node, n_h);
  }

  // 3) alias gather + concat + l2norm
  k_seq_build<<<(B * L * 32 + 255) / 256, 256, 0, stream>>>(n_h, alias_item, alias_cate, seq);

  // 4) attention readout (q2 on the matrix pipe)
  k_q1<<<B, 256, 0, stream>>>(item_seq, seq, q1_w, q1);
  {
    dim3 g2(B * L / 128, D / 64);
    k_q2_wmma<<<g2, 256, 0, stream>>>(seq, q2_w, q2b);
  }
  k_alpha<<<(B * L * 32 + 255) / 256, 256, 0, stream>>>(q2b, q1, q1_b, v_w, alpha);
  k_seqout<<<B, 256, 0, stream>>>(seq, alpha, item_seq, qn);

  // 5) normalized item embedding matrix
  k_item_norm<<<(ITEM_NUM * 32 + 255) / 256, 256, 0, stream>>>(embedding, item_cates, itn);

  // 6) scores GEMM on the matrix pipe (f32 WMMA), scaled by 16
  dim3 grid(B / 64, (ITEM_NUM + 127) / 128);
  k_scores_wmma<<<grid, 256, 0, stream>>>(qn, itn, scores);
}